// Net6_30322469110251
// MI455X (gfx1250) — compile-verified
//
#include <hip/hip_runtime.h>
#include <hip/hip_bf16.h>
#include <stdint.h>

#define NN 100000
#define EE 1200000
#define GG 256
#define CC 64
#define ICN 32

typedef float v2f __attribute__((ext_vector_type(2)));
typedef float v8f __attribute__((ext_vector_type(8)));

// ---------------- utility fills ----------------
__global__ void k_fill_f32(float* __restrict__ p, float v, int n) {
  int i = blockIdx.x * blockDim.x + threadIdx.x;
  if (i < n) p[i] = v;
}
__global__ void k_fill_u32(unsigned* __restrict__ p, unsigned v, int n) {
  int i = blockIdx.x * blockDim.x + threadIdx.x;
  if (i < n) p[i] = v;
}

// ---------------- degree / normalization ----------------
__global__ void k_deg(const int* __restrict__ ei, const float* __restrict__ ea,
                      float* __restrict__ deg1, float* __restrict__ deg2) {
  int e = blockIdx.x * blockDim.x + threadIdx.x;
  if (e >= EE) return;
  int d = ei[EE + e];
  atomicAdd(&deg1[d], 1.0f);     // conv1 uses unit edge weights
  atomicAdd(&deg2[d], ea[e]);    // conv2 uses edge_attr
}

__global__ void k_dinv(float* __restrict__ d1, float* __restrict__ d2) {
  int i = blockIdx.x * blockDim.x + threadIdx.x;
  if (i >= NN) return;
  d1[i] = rsqrtf(d1[i] + 1.0f);  // +1: self-loop weight; deg > 0 always
  d2[i] = rsqrtf(d2[i] + 1.0f);
}

__global__ void k_cnt(const int* __restrict__ batch, float* __restrict__ cnt) {
  int i = blockIdx.x * blockDim.x + threadIdx.x;
  if (i >= NN) return;
  atomicAdd(&cnt[batch[i]], 1.0f);
}

// ---------------- WMMA GEMM: Y[N x 64] = X[N x K] * W[K x 64] ----------------
// fp32 path -> V_WMMA_F32_16X16X4_F32. Block = 128 threads (4 waves).
// blockIdx.x selects the 16-row tile; wave selects the 16-col tile.
// A 16x4 layout (ISA 7.12.2): lane m=lane&15, kh=lane>>4; VGPR0=K(2kh), VGPR1=K(2kh+1).
// B 4x16 mirrored; C/D: VGPR r = rows r (lanes 0-15) / r+8 (lanes 16-31).
template<int K>
__global__ void k_gemm_wmma(const float* __restrict__ X, const float* __restrict__ W,
                            float* __restrict__ Y) {
  const int lane = threadIdx.x & 31;
  const int wave = threadIdx.x >> 5;
  const int m    = lane & 15;
  const int kh   = lane >> 4;
  const int row  = blockIdx.x * 16 + m;   // NN is a multiple of 16: no guards, EXEC all-1s
  const int col  = wave * 16 + m;
  v8f c = {};
  #pragma unroll
  for (int k = 0; k < K; k += 4) {
    v2f a, b;
    a.x = X[(size_t)row * K + k + 2 * kh];
    a.y = X[(size_t)row * K + k + 2 * kh + 1];
    b.x = W[(size_t)(k + 2 * kh) * CC + col];
    b.y = W[(size_t)(k + 2 * kh + 1) * CC + col];
    c = __builtin_amdgcn_wmma_f32_16x16x4_f32(false, a, false, b, (short)0, c,
                                              false, false);
  }
  const int rbase = blockIdx.x * 16 + kh * 8;
  #pragma unroll
  for (int r = 0; r < 8; ++r) {
    Y[(size_t)(rbase + r) * CC + col] = c[r];
  }
}

// ---------------- edge scatter (GCN aggregation) ----------------
// 16 threads per edge, float4 per thread; atomics land in L2 (xw is L2-resident).
__global__ void k_scatter(const int* __restrict__ ei, const float* __restrict__ ea,
                          const float* __restrict__ dinv, const float* __restrict__ xw,
                          float* __restrict__ out, int use_ea) {
  int t = blockIdx.x * blockDim.x + threadIdx.x;
  if (t >= EE * 16) return;
  int e = t >> 4;
  int c = (t & 15) * 4;
  int s = ei[e];
  int d = ei[EE + e];
  __builtin_prefetch(xw + (size_t)s * CC, 0, 1);
  float w = use_ea ? ea[e] : 1.0f;
  float norm = dinv[s] * w * dinv[d];
  const float4 v = *(const float4*)(xw + (size_t)s * CC + c);
  float* o = out + (size_t)d * CC + c;
  atomicAdd(o + 0, v.x * norm);
  atomicAdd(o + 1, v.y * norm);
  atomicAdd(o + 2, v.z * norm);
  atomicAdd(o + 3, v.w * norm);
}

// self-loop contribution + bias (+ optional PReLU)
__global__ void k_finish(const float* __restrict__ xw, const float* __restrict__ dinv,
                         const float* __restrict__ bias, float* __restrict__ out,
                         const float* __restrict__ prelu_w) {
  int i = blockIdx.x * blockDim.x + threadIdx.x;
  if (i >= NN * CC) return;
  int node = i >> 6, c = i & 63;
  float di = dinv[node];
  float v = out[i] + xw[i] * di * di + bias[c];
  if (prelu_w) v = (v >= 0.0f) ? v : prelu_w[c] * v;
  out[i] = v;
}

// ---------------- GraphNorm (two-pass) ----------------
__global__ void k_gsum(const float* __restrict__ h, const int* __restrict__ batch,
                       float* __restrict__ gsum) {
  int i = blockIdx.x * blockDim.x + threadIdx.x;
  if (i >= NN * CC) return;
  int node = i >> 6, c = i & 63;
  atomicAdd(&gsum[batch[node] * CC + c], h[i]);
}

__global__ void k_gcenter(float* __restrict__ h, const int* __restrict__ batch,
                          const float* __restrict__ gsum, const float* __restrict__ cnt,
                          const float* __restrict__ ms, float* __restrict__ gsq) {
  int i = blockIdx.x * blockDim.x + threadIdx.x;
  if (i >= NN * CC) return;
  int node = i >> 6, c = i & 63;
  int g = batch[node];
  float mean = gsum[g * CC + c] / cnt[g];
  float xc = h[i] - mean * ms[c];
  h[i] = xc;
  atomicAdd(&gsq[g * CC + c], xc * xc);
}

__global__ void k_gfinal(float* __restrict__ h, const int* __restrict__ batch,
                         const float* __restrict__ gsq, const float* __restrict__ cnt,
                         const float* __restrict__ w, const float* __restrict__ b,
                         const float* __restrict__ prelu_w) {
  int i = blockIdx.x * blockDim.x + threadIdx.x;
  if (i >= NN * CC) return;
  int node = i >> 6, c = i & 63;
  int g = batch[node];
  float var = gsq[g * CC + c] / cnt[g];
  float v = w[c] * h[i] * rsqrtf(var + 1e-5f) + b[c];
  if (prelu_w) v = (v >= 0.0f) ? v : prelu_w[c] * v;
  h[i] = v;
}

// ---------------- graph pooling ----------------
__device__ __forceinline__ void atomicMaxF32(float* addr, float v) {
  if (v >= 0.0f) atomicMax((int*)addr, __float_as_int(v));
  else           atomicMin((unsigned int*)addr, __float_as_uint(v));
}

__global__ void k_pool(const float* __restrict__ h, const int* __restrict__ batch,
                       float* __restrict__ psum, unsigned* __restrict__ pmax) {
  int i = blockIdx.x * blockDim.x + threadIdx.x;
  if (i >= NN * CC) return;
  int node = i >> 6, c = i & 63;
  int g = batch[node];
  atomicAdd(&psum[g * CC + c], h[i]);
  atomicMaxF32((float*)&pmax[g * CC + c], h[i]);
}

// ---------------- MLP head: [G,128] -> PReLU(lin1) -> lin3 -> [G] ----------------
__global__ void k_mlp(const float* __restrict__ psum, const unsigned* __restrict__ pmax,
                      const float* __restrict__ cnt, const float* __restrict__ l1w,
                      const float* __restrict__ l1b, const float* __restrict__ w2,
                      const float* __restrict__ l3w, const float* __restrict__ l3b,
                      float* __restrict__ out) {
  int g = blockIdx.x * blockDim.x + threadIdx.x;
  if (g >= GG) return;
  float z[2 * CC];
  float invc = 1.0f / cnt[g];
  #pragma unroll
  for (int c = 0; c < CC; ++c) {
    z[c]      = psum[g * CC + c] * invc;
    z[CC + c] = __uint_as_float(pmax[g * CC + c]);
  }
  float acc = l3b[0];
  for (int ic = 0; ic < ICN; ++ic) {
    float a = l1b[ic];
    #pragma unroll 8
    for (int k = 0; k < 2 * CC; ++k) a += z[k] * l1w[k * ICN + ic];
    a = (a >= 0.0f) ? a : w2[ic] * a;
    acc += a * l3w[ic];
  }
  out[g] = acc;
}

extern "C" void kernel_launch(void* const* d_in, const int* in_sizes, int n_in,
                              void* d_out, int out_size, void* d_ws, size_t ws_size,
                              hipStream_t stream) {
  const float* x        = (const float*)d_in[0];
  const float* ea       = (const float*)d_in[1];
  const float* conv1_w  = (const float*)d_in[2];
  const float* conv1_b  = (const float*)d_in[3];
  const float* conv2_w  = (const float*)d_in[4];
  const float* conv2_b  = (const float*)d_in[5];
  const float* weight1  = (const float*)d_in[6];
  const float* weight2  = (const float*)d_in[7];
  const float* bn1_w    = (const float*)d_in[8];
  const float* bn1_b    = (const float*)d_in[9];
  const float* bn1_ms   = (const float*)d_in[10];
  const float* bn2_w    = (const float*)d_in[11];
  const float* bn2_b    = (const float*)d_in[12];
  const float* bn2_ms   = (const float*)d_in[13];
  const float* lin1_w   = (const float*)d_in[14];
  const float* lin1_b   = (const float*)d_in[15];
  const float* lin3_w   = (const float*)d_in[16];
  const float* lin3_b   = (const float*)d_in[17];
  const int*   ei       = (const int*)d_in[18];
  const int*   batch    = (const int*)d_in[19];
  float* out = (float*)d_out;

  // workspace layout
  float* bufA  = (float*)d_ws;             // N x 64  (X@W)
  float* bufB  = bufA + (size_t)NN * CC;   // N x 64  (aggregated features)
  float* dinv1 = bufB + (size_t)NN * CC;   // N
  float* dinv2 = dinv1 + NN;               // N
  float* cnt   = dinv2 + NN;               // G
  float* gsum  = cnt + GG;                 // G x 64
  float* gsq   = gsum + GG * CC;           // G x 64
  float* psum  = gsq + GG * CC;            // G x 64
  unsigned* pmax = (unsigned*)(psum + GG * CC); // G x 64

  const int TB = 256;
  const int nb_node   = (NN + TB - 1) / TB;          // per-node
  const int nb_nc     = (NN * CC + TB - 1) / TB;     // per node-channel
  const int nb_edge   = (EE + TB - 1) / TB;          // per edge
  const int nb_sc     = (EE * 16 + TB - 1) / TB;     // scatter threads
  const int nb_gc     = (GG * CC + TB - 1) / TB;

  // degree / count precompute
  k_fill_f32<<<nb_node, TB, 0, stream>>>(dinv1, 0.0f, NN);
  k_fill_f32<<<nb_node, TB, 0, stream>>>(dinv2, 0.0f, NN);
  k_fill_f32<<<1, TB, 0, stream>>>(cnt, 0.0f, GG);
  k_deg<<<nb_edge, TB, 0, stream>>>(ei, ea, dinv1, dinv2);
  k_dinv<<<nb_node, TB, 0, stream>>>(dinv1, dinv2);
  k_cnt<<<nb_node, TB, 0, stream>>>(batch, cnt);

  // ---- conv1: xw = x @ W1 (WMMA), aggregate, +bias
  k_gemm_wmma<32><<<NN / 16, 128, 0, stream>>>(x, conv1_w, bufA);
  k_fill_f32<<<nb_nc, TB, 0, stream>>>(bufB, 0.0f, NN * CC);
  k_scatter<<<nb_sc, TB, 0, stream>>>(ei, ea, dinv1, bufA, bufB, 0);
  k_finish<<<nb_nc, TB, 0, stream>>>(bufA, dinv1, conv1_b, bufB, nullptr);

  // ---- GraphNorm1 + PReLU(weight1)
  k_fill_f32<<<nb_gc, TB, 0, stream>>>(gsum, 0.0f, GG * CC);
  k_gsum<<<nb_nc, TB, 0, stream>>>(bufB, batch, gsum);
  k_fill_f32<<<nb_gc, TB, 0, stream>>>(gsq, 0.0f, GG * CC);
  k_gcenter<<<nb_nc, TB, 0, stream>>>(bufB, batch, gsum, cnt, bn1_ms, gsq);
  k_gfinal<<<nb_nc, TB, 0, stream>>>(bufB, batch, gsq, cnt, bn1_w, bn1_b, weight1);

  // ---- conv2: xw = h @ W2 (WMMA), weighted aggregate, +bias, PReLU(weight1)
  k_gemm_wmma<64><<<NN / 16, 128, 0, stream>>>(bufB, conv2_w, bufA);
  k_fill_f32<<<nb_nc, TB, 0, stream>>>(bufB, 0.0f, NN * CC);
  k_scatter<<<nb_sc, TB, 0, stream>>>(ei, ea, dinv2, bufA, bufB, 1);
  k_finish<<<nb_nc, TB, 0, stream>>>(bufA, dinv2, conv2_b, bufB, weight1);

  // ---- GraphNorm2 (no PReLU after)
  k_fill_f32<<<nb_gc, TB, 0, stream>>>(gsum, 0.0f, GG * CC);
  k_gsum<<<nb_nc, TB, 0, stream>>>(bufB, batch, gsum);
  k_fill_f32<<<nb_gc, TB, 0, stream>>>(gsq, 0.0f, GG * CC);
  k_gcenter<<<nb_nc, TB, 0, stream>>>(bufB, batch, gsum, cnt, bn2_ms, gsq);
  k_gfinal<<<nb_nc, TB, 0, stream>>>(bufB, batch, gsq, cnt, bn2_w, bn2_b, nullptr);

  // ---- pooling (mean + max) and MLP head
  k_fill_f32<<<nb_gc, TB, 0, stream>>>(psum, 0.0f, GG * CC);
  k_fill_u32<<<nb_gc, TB, 0, stream>>>(pmax, 0xFF800000u, GG * CC); // -inf
  k_pool<<<nb_nc, TB, 0, stream>>>(bufB, batch, psum, pmax);
  k_mlp<<<1, TB, 0, stream>>>(psum, pmax, cnt, lin1_w, lin1_b, weight2,
                              lin3_w, lin3_b, out);

  (void)in_sizes; (void)n_in; (void)out_size; (void)ws_size;
}